// ResGraphConv_59588376264957
// MI455X (gfx1250) — compile-verified
//
#include <hip/hip_runtime.h>
#include <math.h>

typedef float v2f __attribute__((ext_vector_type(2)));
typedef float v8f __attribute__((ext_vector_type(8)));

// ---------------------------------------------------------------------------
// Float atomic max via signed/unsigned integer ordering trick (works with
// -inf init; mixed-sign safe).
// ---------------------------------------------------------------------------
__device__ __forceinline__ void atomicMaxF(float* addr, float val) {
    if (val >= 0.0f)
        atomicMax((int*)addr, __float_as_int(val));
    else
        atomicMin((unsigned int*)addr, __float_as_uint(val));
}

// ---------------------------------------------------------------------------
// Y[nrows, ncols] = X[nrows, 64] @ W[64, ncols] + bias (+ addend rows)
// One block = one 16-row tile of X staged in LDS; 8 waves sweep 16-col tiles.
// Each wave computes a 16x16 f32 tile with 16x V_WMMA_F32_16X16X4_F32 (K=64).
// A-frag (16x4 f32): lanes 0-15 -> M, v0/v1 = K0/K1; lanes 16-31 -> K2/K3.
// B-frag (4x16 f32): mirrored striping; D: VGPR i = rows {i, i+8}, N = lane%16.
// Loads are hoisted into register arrays so all 32 global loads issue before
// the WMMAs, which then drain with partial s_wait_loadcnt (pipelined).
// ---------------------------------------------------------------------------
__global__ __launch_bounds__(256) void gemm_xw_wmma_kernel(
    const float* __restrict__ X, const float* __restrict__ W,
    const float* __restrict__ bias, const float* __restrict__ addend,
    float* __restrict__ Y, int nrows, int ncols)
{
    __shared__ float xs[16][68];   // pad 68: 16 lanes of one K hit distinct banks
    const int row0 = blockIdx.x * 16;
    const bool fullTile = (row0 + 16 <= nrows);

    for (int i = threadIdx.x; i < 16 * 64; i += 256) {
        int r = i >> 6, c = i & 63;
        int gr = row0 + r;
        xs[r][c] = (fullTile || gr < nrows) ? X[(size_t)gr * 64 + c] : 0.0f;
    }
    __syncthreads();

    const int wave  = threadIdx.x >> 5;
    const int lane  = threadIdx.x & 31;
    const int r     = lane & 15;             // M (A) / N (B,D) index
    const int halfk = (lane >> 4) << 1;      // 0 for lanes 0-15, 2 for 16-31
    const int ntiles = ncols >> 4;

    // A fragments: xs[r][4j+halfk .. +1] -> 8-byte aligned LDS pair reads.
    v2f afrag[16];
    {
        const v2f* xp = (const v2f*)(&xs[r][halfk]);   // 8B aligned (halfk even)
#pragma unroll
        for (int j = 0; j < 16; ++j)
            afrag[j] = xp[2 * j];
    }

    for (int ct = wave; ct < ntiles; ct += 8) {
        const int col = (ct << 4) + r;

        // ---- load phase: 32 independent global loads ----
        v2f bfrag[16];
        {
            const float* wp = W + (size_t)halfk * ncols + col;
#pragma unroll
            for (int j = 0; j < 16; ++j) {
                bfrag[j].x = wp[(size_t)(4 * j)     * ncols];
                bfrag[j].y = wp[(size_t)(4 * j + 1) * ncols];
            }
        }

        // ---- compute phase: 16 chained WMMAs ----
        v8f acc = {0.f, 0.f, 0.f, 0.f, 0.f, 0.f, 0.f, 0.f};
#pragma unroll
        for (int j = 0; j < 16; ++j)
            acc = __builtin_amdgcn_wmma_f32_16x16x4_f32(
                false, afrag[j], false, bfrag[j], (short)0, acc, false, false);

        // ---- epilogue: guard-free fast path for full row tiles ----
        const float bb = bias[col];
        const int mbase = (lane < 16) ? 0 : 8;
        float* yp = Y + (size_t)(row0 + mbase) * ncols + col;
        if (fullTile) {
            if (addend) {                      // skip GEMM path (ncols == 64)
                const float* ap = addend + (size_t)(row0 + mbase) * 64 + col;
#pragma unroll
                for (int i = 0; i < 8; ++i)
                    yp[(size_t)i * ncols] = acc[i] + bb + ap[(size_t)i * 64];
            } else {
#pragma unroll
                for (int i = 0; i < 8; ++i)
                    yp[(size_t)i * ncols] = acc[i] + bb;
            }
        } else {
#pragma unroll
            for (int i = 0; i < 8; ++i) {
                int gr = row0 + mbase + i;
                if (gr < nrows) {
                    float val = acc[i] + bb;
                    if (addend) val += addend[(size_t)gr * 64 + col];
                    Y[(size_t)gr * ncols + col] = val;
                }
            }
        }
    }
}

// ---------------------------------------------------------------------------
__global__ void init_md_kernel(float* __restrict__ m, float* __restrict__ denom, int n)
{
    int i = blockIdx.x * blockDim.x + threadIdx.x;
    if (i < n) { m[i] = -INFINITY; denom[i] = 0.0f; }
}

// alpha[e,h] = dot(q[dst[e],h,:], k[src[e],h,:]) / 8 ; segment max into mbuf
__global__ __launch_bounds__(256) void alpha_kernel(
    const float* __restrict__ q, const float* __restrict__ k,
    const int* __restrict__ src, const int* __restrict__ dst,
    float* __restrict__ alpha, float* __restrict__ mbuf, int E)
{
    int idx = blockIdx.x * blockDim.x + threadIdx.x;   // e*8 + h
    if (idx >= E * 8) return;
    int e = idx >> 3, h = idx & 7;
    int d = dst[e], s = src[e];
    const float4* qp = (const float4*)(q + (size_t)d * 512 + h * 64);
    const float4* kp = (const float4*)(k + (size_t)s * 512 + h * 64);
    float acc = 0.0f;
#pragma unroll
    for (int i = 0; i < 16; ++i) {
        float4 a = qp[i], b = kp[i];
        acc += a.x * b.x + a.y * b.y + a.z * b.z + a.w * b.w;
    }
    acc *= 0.125f;                 // 1/sqrt(C), C=64
    alpha[idx] = acc;
    atomicMaxF(&mbuf[d * 8 + h], acc);
}

// ealpha = exp(alpha - m[dst]) in place; denom[dst,h] += ealpha
__global__ __launch_bounds__(256) void exp_kernel(
    float* __restrict__ alpha, const float* __restrict__ mbuf,
    const int* __restrict__ dst, float* __restrict__ denom, int E)
{
    int idx = blockIdx.x * blockDim.x + threadIdx.x;
    if (idx >= E * 8) return;
    int e = idx >> 3, h = idx & 7;
    int d = dst[e];
    float v = __expf(alpha[idx] - mbuf[d * 8 + h]);
    alpha[idx] = v;
    atomicAdd(&denom[d * 8 + h], v);
}

// out[dst, c] += sum_h a[e,h] * v[src,h,c] / H   (head mean folded into a)
// One wave per edge; lane covers c = {lane, lane+32}.
__global__ __launch_bounds__(256) void scatter_kernel(
    const float* __restrict__ vmat, const float* __restrict__ ealpha,
    const float* __restrict__ denom, const int* __restrict__ src,
    const int* __restrict__ dst, float* __restrict__ out, int E)
{
    int wave = threadIdx.x >> 5, lane = threadIdx.x & 31;
    int e = blockIdx.x * 8 + wave;
    if (e >= E) return;
    int s = src[e], d = dst[e];
    float a[8];
#pragma unroll
    for (int h = 0; h < 8; ++h)
        a[h] = ealpha[(size_t)e * 8 + h] /
               (denom[(size_t)d * 8 + h] + 1e-16f) * 0.125f;   // * 1/H
    const float* vp = vmat + (size_t)s * 512;
    float acc0 = 0.0f, acc1 = 0.0f;
#pragma unroll
    for (int h = 0; h < 8; ++h) {
        acc0 += a[h] * vp[h * 64 + lane];
        acc1 += a[h] * vp[h * 64 + 32 + lane];
    }
    atomicAdd(&out[(size_t)d * 64 + lane], acc0);
    atomicAdd(&out[(size_t)d * 64 + 32 + lane], acc1);
}

// ---------------------------------------------------------------------------
extern "C" void kernel_launch(void* const* d_in, const int* in_sizes, int n_in,
                              void* d_out, int out_size, void* d_ws, size_t ws_size,
                              hipStream_t stream)
{
    const float* x   = (const float*)d_in[0];
    const int*   ei  = (const int*)  d_in[1];
    const float* Wq  = (const float*)d_in[2];
    const float* bq  = (const float*)d_in[3];
    const float* Wk  = (const float*)d_in[4];
    const float* bk  = (const float*)d_in[5];
    const float* Wv  = (const float*)d_in[6];
    const float* bv  = (const float*)d_in[7];
    const float* Wsk = (const float*)d_in[8];
    const float* bsk = (const float*)d_in[9];
    float* out = (float*)d_out;

    const int N  = in_sizes[0] / 64;     // 50000
    const int E  = in_sizes[1] / 2;      // 200000
    const int HC = in_sizes[3];          // 512
    const int* src = ei;
    const int* dst = ei + E;

    float* ws = (float*)d_ws;
    size_t off = 0;
    float* q  = ws + off; off += (size_t)N * HC;
    float* k  = ws + off; off += (size_t)N * HC;
    float* v  = ws + off; off += (size_t)N * HC;
    float* al = ws + off; off += (size_t)E * 8;
    float* m  = ws + off; off += (size_t)N * 8;
    float* dn = ws + off; off += (size_t)N * 8;

    const int gRows = (N + 15) / 16;
    gemm_xw_wmma_kernel<<<gRows, 256, 0, stream>>>(x, Wq,  bq,  nullptr, q,   N, HC);
    gemm_xw_wmma_kernel<<<gRows, 256, 0, stream>>>(x, Wk,  bk,  nullptr, k,   N, HC);
    gemm_xw_wmma_kernel<<<gRows, 256, 0, stream>>>(x, Wv,  bv,  nullptr, v,   N, HC);
    // out = x + x@Wskip + bskip   (residual + root weight, scatter adds later)
    gemm_xw_wmma_kernel<<<gRows, 256, 0, stream>>>(x, Wsk, bsk, x,       out, N, 64);

    init_md_kernel<<<(N * 8 + 255) / 256, 256, 0, stream>>>(m, dn, N * 8);
    alpha_kernel  <<<(E * 8 + 255) / 256, 256, 0, stream>>>(q, k, src, dst, al, m, E);
    exp_kernel    <<<(E * 8 + 255) / 256, 256, 0, stream>>>(al, m, dst, dn, E);
    scatter_kernel<<<(E + 7) / 8,         256, 0, stream>>>(v, al, dn, src, dst, out, E);
}